// _CycleDartTokenizer_46815143526785
// MI455X (gfx1250) — compile-verified
//
#include <hip/hip_runtime.h>
#include <hip/hip_bf16.h>

#define LCYC 256
#define WDIM 8
#define NCYC 1024
#define CLAMPV 10000.0f

typedef float v2f __attribute__((ext_vector_type(2)));
typedef float v8f __attribute__((ext_vector_type(8)));

__device__ __forceinline__ float safev(float x) {
    x = (x == x) ? x : 0.0f;          // NaN -> 0
    x = fminf(x, CLAMPV);             // +inf -> CLAMP
    x = fmaxf(x, -CLAMPV);            // -inf -> -CLAMP
    return x;
}

__device__ __forceinline__ void ln8(const float* x, const float* g, const float* b, float* y) {
    float mu = 0.0f;
#pragma unroll
    for (int j = 0; j < 8; ++j) mu += x[j];
    mu *= 0.125f;
    float var = 0.0f;
#pragma unroll
    for (int j = 0; j < 8; ++j) { float d = x[j] - mu; var += d * d; }
    var *= 0.125f;
    float r = rsqrtf(var + 1e-5f);
#pragma unroll
    for (int j = 0; j < 8; ++j) y[j] = (x[j] - mu) * r * g[j] + b[j];
}

__device__ __forceinline__ float wave_sum(float v) {
#pragma unroll
    for (int d = 16; d > 0; d >>= 1) v += __shfl_xor(v, d);
    return v;
}
__device__ __forceinline__ float wave_max(float v) {
#pragma unroll
    for (int d = 16; d > 0; d >>= 1) v = fmaxf(v, __shfl_xor(v, d));
    return v;
}

// ---------------------------------------------------------------------------
// Kernel 1: one block per cycle. Full per-cycle pipeline, incl. stage-2 QKV.
// ---------------------------------------------------------------------------
__global__ __launch_bounds__(256) void k_cycle(
    const float* __restrict__ df,
    const float* __restrict__ de_w, const float* __restrict__ de_b,
    const float* __restrict__ ca_in_w, const float* __restrict__ ca_in_b,
    const float* __restrict__ ca_out_w, const float* __restrict__ ca_out_b,
    const float* __restrict__ cn_g, const float* __restrict__ cn_b,
    const float* __restrict__ cs_w, const float* __restrict__ cs_b,
    const float* __restrict__ cp_ln_g, const float* __restrict__ cp_ln_b,
    const float* __restrict__ cp_w1, const float* __restrict__ cp_b1,
    const float* __restrict__ cp_w2, const float* __restrict__ cp_b2,
    const float* __restrict__ ia_in_w, const float* __restrict__ ia_in_b,
    float* __restrict__ ws_cyc, float* __restrict__ ws_qkv2)
{
    __shared__ float s_ox[LCYC], s_oy[LCYC];     // original darts
    __shared__ float s_vx[LCYC], s_vy[LCYC];     // inclusive scan
    __shared__ float s_emb[LCYC][WDIM + 1];      // padded
    __shared__ float s_qkv[LCYC][33];            // 24 used, cols 24..31 scratch
    __shared__ float s_w8x[8], s_w8y[8];         // per-wave scan totals
    __shared__ float s_part[8];                  // per-wave reduction partials
    __shared__ float s_pool[8][9];               // per-wave pooled partial vectors
    __shared__ float s_bc[8];                    // broadcast cyc

    const int tid  = threadIdx.x;
    const int lane = tid & 31;
    const int wave = tid >> 5;
    const int c    = blockIdx.x;
    const float* base = df + (size_t)c * LCYC * 3;

    // ---- load darts (only xy used by reference) ----
    float dx = base[tid * 3 + 0];
    float dy = base[tid * 3 + 1];
    s_ox[tid] = dx; s_oy[tid] = dy;

    // ---- inclusive scan: intra-wave shuffle scan + cross-wave offsets ----
    float vx = dx, vy = dy;
#pragma unroll
    for (int d = 1; d < 32; d <<= 1) {
        float tx = __shfl_up(vx, d);
        float ty = __shfl_up(vy, d);
        if (lane >= d) { vx += tx; vy += ty; }
    }
    if (lane == 31) { s_w8x[wave] = vx; s_w8y[wave] = vy; }
    __syncthreads();                                                   // B1
    float offx = 0.0f, offy = 0.0f;
#pragma unroll
    for (int w = 0; w < 8; ++w) {
        float px = s_w8x[w], py = s_w8y[w];
        if (w < wave) { offx += px; offy += py; }
    }
    vx += offx; vy += offy;                 // inclusive sum at tid
    s_vx[tid] = vx; s_vy[tid] = vy;
    __syncthreads();                                                   // B2

    // ---- shoelace area: verts[i] = exclusive sum, V(i+1) = inclusive(i) ----
    float xi  = (tid == 0) ? 0.0f : s_vx[tid - 1];
    float yi  = (tid == 0) ? 0.0f : s_vy[tid - 1];
    float xip = (tid == LCYC - 1) ? 0.0f : vx;      // V(tid+1)
    float yip = (tid == LCYC - 1) ? 0.0f : vy;
    float term = wave_sum(xi * yip - yi * xip);
    if (lane == 0) s_part[wave] = term;
    __syncthreads();                                                   // B3
    float area = 0.0f;
#pragma unroll
    for (int w = 0; w < 8; ++w) area += s_part[w];
    area *= 0.5f;

    // ---- orientation normalize ----
    float x, y;
    if (area >= 0.0f) { x = s_ox[tid];             y = s_oy[tid]; }
    else              { x = -s_ox[LCYC - 1 - tid]; y = -s_oy[LCYC - 1 - tid]; }

    // ---- feats + dart embed (ReLU) ----
    float f0 = safev(x), f1 = safev(y), f2 = safev(sqrtf(x * x + y * y));
    float emb[WDIM];
#pragma unroll
    for (int w = 0; w < WDIM; ++w) {
        float t = f0 * de_w[w * 3 + 0] + f1 * de_w[w * 3 + 1] + f2 * de_w[w * 3 + 2] + de_b[w];
        emb[w] = safev(fmaxf(t, 0.0f));
        s_emb[tid][w] = emb[w];
    }
    __syncthreads();                                                   // B4

    // ---- QKV projection: (256 x 8) @ (8 x 24) via WMMA f32 16x16x4 ----
#if defined(__gfx1250__) && __has_builtin(__builtin_amdgcn_wmma_f32_16x16x4_f32)
    {
        const int lm = lane & 15;
        const int hi = lane >> 4;        // 0: K0/K1 half, 1: K2/K3 half
        // Hoist B fragments + bias (invariant across row tiles); branchless.
        v2f bfrag[2][2];
        float bias[2];
#pragma unroll
        for (int nt = 0; nt < 2; ++nt) {
            const int n   = nt * 16 + lm;
            const int ncl = (n < 24) ? n : 0;           // clamped, always-valid address
            const bool ok = (n < 24);
            float bb = ca_in_b[ncl];
            bias[nt] = ok ? bb : 0.0f;
#pragma unroll
            for (int kc = 0; kc < 2; ++kc) {
                const int kb = hi * 2 + kc * 4;
                float b0 = ca_in_w[ncl * 8 + kb];
                float b1 = ca_in_w[ncl * 8 + kb + 1];
                bfrag[nt][kc].x = ok ? b0 : 0.0f;
                bfrag[nt][kc].y = ok ? b1 : 0.0f;
            }
        }
#pragma unroll
        for (int mt = 0; mt < 2; ++mt) {
            const int mrow = (wave * 2 + mt) * 16 + lm;  // A row for this lane
            v2f a0, a1;
            a0.x = s_emb[mrow][hi * 2];     a0.y = s_emb[mrow][hi * 2 + 1];
            a1.x = s_emb[mrow][hi * 2 + 4]; a1.y = s_emb[mrow][hi * 2 + 5];
#pragma unroll
            for (int nt = 0; nt < 2; ++nt) {
                v8f acc;
#pragma unroll
                for (int r = 0; r < 8; ++r) acc[r] = bias[nt];
                acc = __builtin_amdgcn_wmma_f32_16x16x4_f32(
                    false, a0, false, bfrag[nt][0], (short)0, acc, false, false);
                acc = __builtin_amdgcn_wmma_f32_16x16x4_f32(
                    false, a1, false, bfrag[nt][1], (short)0, acc, false, false);
                const int rowb = (wave * 2 + mt) * 16 + hi * 8;
                const int col  = nt * 16 + lm;           // 0..31, unconditional store
#pragma unroll
                for (int r = 0; r < 8; ++r) s_qkv[rowb + r][col] = acc[r];
            }
        }
    }
#else
    {
        for (int o = 0; o < 24; ++o) {
            float t = ca_in_b[o];
#pragma unroll
            for (int j = 0; j < 8; ++j) t += s_emb[tid][j] * ca_in_w[o * 8 + j];
            s_qkv[tid][o] = t;
        }
    }
#endif
    __syncthreads();                                                   // B5

    // ---- cycle attention: mask allows only {i-1, i, i+1} (others underflow) ----
    const int im = (tid + LCYC - 1) & (LCYC - 1);
    const int ip = (tid + 1) & (LCYC - 1);
    float q[8], ao[8];
#pragma unroll
    for (int j = 0; j < 8; ++j) q[j] = s_qkv[tid][j];
#pragma unroll
    for (int h = 0; h < 2; ++h) {
        const int b4 = h * 4;
        float sp = 0.0f, ss = 0.0f, sn = 0.0f;
#pragma unroll
        for (int j = 0; j < 4; ++j) {
            float qj = q[b4 + j];
            sp += qj * s_qkv[im][8 + b4 + j];
            ss += qj * s_qkv[tid][8 + b4 + j];
            sn += qj * s_qkv[ip][8 + b4 + j];
        }
        sp *= 0.5f; ss *= 0.5f; sn *= 0.5f;   // 1/sqrt(d), d=4
        float mx = fmaxf(sp, fmaxf(ss, sn));
        float ep = __expf(sp - mx), es = __expf(ss - mx), en = __expf(sn - mx);
        float inv = 1.0f / (ep + es + en);
        ep *= inv; es *= inv; en *= inv;
#pragma unroll
        for (int j = 0; j < 4; ++j)
            ao[b4 + j] = ep * s_qkv[im][16 + b4 + j]
                       + es * s_qkv[tid][16 + b4 + j]
                       + en * s_qkv[ip][16 + b4 + j];
    }

    // ---- out proj + residual + LN ----
    float x2[8], e2[8];
#pragma unroll
    for (int w = 0; w < 8; ++w) {
        float t = ca_out_b[w];
#pragma unroll
        for (int j = 0; j < 8; ++j) t += ao[j] * ca_out_w[w * 8 + j];
        x2[w] = safev(emb[w] + t);
    }
    ln8(x2, cn_g, cn_b, e2);
#pragma unroll
    for (int w = 0; w < 8; ++w) e2[w] = safev(e2[w]);

    // ---- per-cycle softmax pooling (shuffle reductions) ----
    float lgv = cs_b[0];
#pragma unroll
    for (int j = 0; j < 8; ++j) lgv += e2[j] * cs_w[j];
    lgv = safev(lgv);

    float wm = wave_max(lgv);
    if (lane == 0) s_part[wave] = wm;
    __syncthreads();                                                   // B6
    float mx = -1e30f;
#pragma unroll
    for (int w = 0; w < 8; ++w) mx = fmaxf(mx, s_part[w]);
    float ev = __expf(lgv - mx);
    float wsum = wave_sum(ev);
    __syncthreads();                                                   // B7 (protect s_part)
    if (lane == 0) s_part[wave] = wsum;
    __syncthreads();                                                   // B8
    float Z = 0.0f;
#pragma unroll
    for (int w = 0; w < 8; ++w) Z += s_part[w];
    float wgt = ev / fmaxf(Z, 1e-6f);

    // pooled[j] = sum_i e2[i][j] * wgt_i  (per-wave shuffle reduce, 8 partials)
    float pv[8];
#pragma unroll
    for (int j = 0; j < 8; ++j) pv[j] = wave_sum(e2[j] * wgt);
    if (lane == 0) {
#pragma unroll
        for (int j = 0; j < 8; ++j) s_pool[wave][j] = pv[j];
    }
    __syncthreads();                                                   // B9

    if (tid == 0) {
        float pooled[8], hp[8], h1[8];
#pragma unroll
        for (int j = 0; j < 8; ++j) {
            float t = 0.0f;
#pragma unroll
            for (int w = 0; w < 8; ++w) t += s_pool[w][j];
            pooled[j] = t;
        }
        ln8(pooled, cp_ln_g, cp_ln_b, hp);
#pragma unroll
        for (int w = 0; w < 8; ++w) {
            float t = cp_b1[w];
#pragma unroll
            for (int j = 0; j < 8; ++j) t += hp[j] * cp_w1[w * 8 + j];
            h1[w] = fmaxf(t, 0.0f);
        }
#pragma unroll
        for (int w = 0; w < 8; ++w) {
            float t = cp_b2[w];
#pragma unroll
            for (int j = 0; j < 8; ++j) t += h1[j] * cp_w2[w * 8 + j];
            t = safev(t);
            s_bc[w] = t;
            ws_cyc[c * 8 + w] = t;
        }
    }
    __syncthreads();                                                   // B10
    // stage-2 QKV for this token, spread over 24 threads
    if (tid < 24) {
        float t = ia_in_b[tid];
#pragma unroll
        for (int j = 0; j < 8; ++j) t += s_bc[j] * ia_in_w[tid * 8 + j];
        ws_qkv2[c * 24 + tid] = t;
    }
}

// ---------------------------------------------------------------------------
// Kernel 2: inter-cycle full attention (1024 tokens, 2 heads, d=4),
// online softmax per row; out-proj + residual + LN + stage-2 logits.
// ---------------------------------------------------------------------------
__global__ __launch_bounds__(256) void k_inter(
    const float* __restrict__ ws_cyc, const float* __restrict__ ws_qkv2,
    const float* __restrict__ ia_out_w, const float* __restrict__ ia_out_b,
    const float* __restrict__ in_g, const float* __restrict__ in_b,
    const float* __restrict__ is_w, const float* __restrict__ is_b,
    float* __restrict__ ws_coll, float* __restrict__ ws_lg)
{
    __shared__ float s_kv[NCYC][16];   // k: [0..7], v: [8..15]  (64 KB)
    const int tid = threadIdx.x;
    for (int t = tid; t < NCYC; t += 256) {
#pragma unroll
        for (int j = 0; j < 16; ++j) s_kv[t][j] = ws_qkv2[t * 24 + 8 + j];
    }
    __syncthreads();

    const int i = blockIdx.x * 256 + tid;
    float q[8];
#pragma unroll
    for (int j = 0; j < 8; ++j) q[j] = ws_qkv2[i * 24 + j];

    float m0 = -1e30f, m1 = -1e30f, Z0 = 0.0f, Z1 = 0.0f;
    float acc[8];
#pragma unroll
    for (int j = 0; j < 8; ++j) acc[j] = 0.0f;

    for (int t = 0; t < NCYC; ++t) {
        float s0 = 0.0f, s1 = 0.0f;
#pragma unroll
        for (int j = 0; j < 4; ++j) {
            s0 += q[j]     * s_kv[t][j];
            s1 += q[4 + j] * s_kv[t][4 + j];
        }
        s0 *= 0.5f; s1 *= 0.5f;
        {
            float nm = fmaxf(m0, s0);
            float sc = __expf(m0 - nm), e = __expf(s0 - nm);
            Z0 = Z0 * sc + e;
#pragma unroll
            for (int j = 0; j < 4; ++j) acc[j] = acc[j] * sc + e * s_kv[t][8 + j];
            m0 = nm;
        }
        {
            float nm = fmaxf(m1, s1);
            float sc = __expf(m1 - nm), e = __expf(s1 - nm);
            Z1 = Z1 * sc + e;
#pragma unroll
            for (int j = 0; j < 4; ++j) acc[4 + j] = acc[4 + j] * sc + e * s_kv[t][12 + j];
            m1 = nm;
        }
    }
    float i0 = 1.0f / Z0, i1 = 1.0f / Z1;
#pragma unroll
    for (int j = 0; j < 4; ++j) { acc[j] *= i0; acc[4 + j] *= i1; }

    float xx[8], cc[8];
#pragma unroll
    for (int w = 0; w < 8; ++w) {
        float t = ia_out_b[w];
#pragma unroll
        for (int j = 0; j < 8; ++j) t += acc[j] * ia_out_w[w * 8 + j];
        xx[w] = safev(ws_cyc[i * 8 + w] + t);
    }
    ln8(xx, in_g, in_b, cc);
    float lgv = is_b[0];
#pragma unroll
    for (int w = 0; w < 8; ++w) {
        cc[w] = safev(cc[w]);
        ws_coll[i * 8 + w] = cc[w];
        lgv += cc[w] * is_w[w];
    }
    ws_lg[i] = safev(lgv);
}

// ---------------------------------------------------------------------------
// Kernel 3: softmax pooling over 1024 tokens + final MLP -> 8 outputs.
// ---------------------------------------------------------------------------
__global__ __launch_bounds__(1024) void k_final(
    const float* __restrict__ ws_coll, const float* __restrict__ ws_lg,
    const float* __restrict__ tp_ln_g, const float* __restrict__ tp_ln_b,
    const float* __restrict__ tp_w1, const float* __restrict__ tp_b1,
    const float* __restrict__ tp_w2, const float* __restrict__ tp_b2,
    float* __restrict__ out)
{
    __shared__ float s_part[32];
    __shared__ float s_pool[32][9];
    const int tid  = threadIdx.x;
    const int lane = tid & 31;
    const int wave = tid >> 5;

    float lg = ws_lg[tid];
    float wm = wave_max(lg);
    if (lane == 0) s_part[wave] = wm;
    __syncthreads();
    float mx = -1e30f;
#pragma unroll
    for (int w = 0; w < 32; ++w) mx = fmaxf(mx, s_part[w]);
    float ev = __expf(lg - mx);
    float wsum = wave_sum(ev);
    __syncthreads();
    if (lane == 0) s_part[wave] = wsum;
    __syncthreads();
    float Z = 0.0f;
#pragma unroll
    for (int w = 0; w < 32; ++w) Z += s_part[w];
    float wgt = ev / fmaxf(Z, 1e-6f);

    float cv[8];
#pragma unroll
    for (int j = 0; j < 8; ++j) cv[j] = wave_sum(ws_coll[tid * 8 + j] * wgt);
    if (lane == 0) {
#pragma unroll
        for (int j = 0; j < 8; ++j) s_pool[wave][j] = cv[j];
    }
    __syncthreads();
    if (tid < 8) {
        float acc = 0.0f;
#pragma unroll
        for (int w = 0; w < 32; ++w) acc += s_pool[w][tid];
        s_pool[0][tid] = acc;   // column-disjoint writes, no race
    }
    __syncthreads();

    if (tid == 0) {
        float ti[8], h2[8], h1[8];
#pragma unroll
        for (int j = 0; j < 8; ++j) ti[j] = s_pool[0][j];
        ln8(ti, tp_ln_g, tp_ln_b, h2);
#pragma unroll
        for (int w = 0; w < 8; ++w) {
            float t = tp_b1[w];
#pragma unroll
            for (int j = 0; j < 8; ++j) t += h2[j] * tp_w1[w * 8 + j];
            h1[w] = fmaxf(t, 0.0f);
        }
#pragma unroll
        for (int w = 0; w < 8; ++w) {
            float t = tp_b2[w];
#pragma unroll
            for (int j = 0; j < 8; ++j) t += h1[j] * tp_w2[w * 8 + j];
            out[w] = safev(t);
        }
    }
}

extern "C" void kernel_launch(void* const* d_in, const int* in_sizes, int n_in,
                              void* d_out, int out_size, void* d_ws, size_t ws_size,
                              hipStream_t stream) {
    const float* df       = (const float*)d_in[0];
    const float* de_w     = (const float*)d_in[1];
    const float* de_b     = (const float*)d_in[2];
    const float* ca_in_w  = (const float*)d_in[3];
    const float* ca_in_b  = (const float*)d_in[4];
    const float* ca_out_w = (const float*)d_in[5];
    const float* ca_out_b = (const float*)d_in[6];
    const float* cn_g     = (const float*)d_in[7];
    const float* cn_b     = (const float*)d_in[8];
    const float* cs_w     = (const float*)d_in[9];
    const float* cs_b     = (const float*)d_in[10];
    const float* cp_ln_g  = (const float*)d_in[11];
    const float* cp_ln_b  = (const float*)d_in[12];
    const float* cp_w1    = (const float*)d_in[13];
    const float* cp_b1    = (const float*)d_in[14];
    const float* cp_w2    = (const float*)d_in[15];
    const float* cp_b2    = (const float*)d_in[16];
    const float* ia_in_w  = (const float*)d_in[17];
    const float* ia_in_b  = (const float*)d_in[18];
    const float* ia_out_w = (const float*)d_in[19];
    const float* ia_out_b = (const float*)d_in[20];
    const float* in_g     = (const float*)d_in[21];
    const float* in_b     = (const float*)d_in[22];
    const float* is_w     = (const float*)d_in[23];
    const float* is_b     = (const float*)d_in[24];
    const float* tp_ln_g  = (const float*)d_in[25];
    const float* tp_ln_b  = (const float*)d_in[26];
    const float* tp_w1    = (const float*)d_in[27];
    const float* tp_b1    = (const float*)d_in[28];
    const float* tp_w2    = (const float*)d_in[29];
    const float* tp_b2    = (const float*)d_in[30];

    float* ws      = (float*)d_ws;
    float* ws_cyc  = ws;                         // 1024*8
    float* ws_qkv2 = ws_cyc + NCYC * 8;          // 1024*24
    float* ws_coll = ws_qkv2 + NCYC * 24;        // 1024*8
    float* ws_lg   = ws_coll + NCYC * 8;         // 1024

    k_cycle<<<NCYC, 256, 0, stream>>>(df, de_w, de_b, ca_in_w, ca_in_b, ca_out_w, ca_out_b,
                                      cn_g, cn_b, cs_w, cs_b, cp_ln_g, cp_ln_b,
                                      cp_w1, cp_b1, cp_w2, cp_b2, ia_in_w, ia_in_b,
                                      ws_cyc, ws_qkv2);
    k_inter<<<4, 256, 0, stream>>>(ws_cyc, ws_qkv2, ia_out_w, ia_out_b,
                                   in_g, in_b, is_w, is_b, ws_coll, ws_lg);
    k_final<<<1, 1024, 0, stream>>>(ws_coll, ws_lg, tp_ln_g, tp_ln_b,
                                    tp_w1, tp_b1, tp_w2, tp_b2, (float*)d_out);
}